// MultiHeadCRA_80152679678666
// MI455X (gfx1250) — compile-verified
//
#include <hip/hip_runtime.h>
#include <hip/hip_bf16.h>

typedef __attribute__((ext_vector_type(2))) float v2f;
typedef __attribute__((ext_vector_type(8))) float v8f;

#define HEADS 8
#define HDIM  128
#define RED   8
#define BATCH 8
#define CCH   1024          // HEADS*HDIM
#define SPA   4096          // 64*64

// ---------------------------------------------------------------------------
// 1) avg[b,c] = mean_s x[b,c,s]   (one block per (b,c) row; float4 streaming)
// ---------------------------------------------------------------------------
__global__ void avg_kernel(const float* __restrict__ x, float* __restrict__ avg) {
    const int row = blockIdx.x;                       // b*CCH + c
    const float4* xr = (const float4*)(x + (size_t)row * SPA);
    float s = 0.f;
    for (int i = threadIdx.x; i < SPA / 4; i += 256) {
        float4 v = xr[i];
        s += v.x + v.y + v.z + v.w;
    }
    for (int off = 16; off > 0; off >>= 1) s += __shfl_xor(s, off, 32);
    __shared__ float wsum[8];
    const int wid = threadIdx.x >> 5, lane = threadIdx.x & 31;
    if (lane == 0) wsum[wid] = s;
    __syncthreads();
    if (threadIdx.x == 0) {
        float t = 0.f;
        for (int i = 0; i < 8; ++i) t += wsum[i];
        avg[row] = t * (1.0f / SPA);
    }
}

// ---------------------------------------------------------------------------
// 2) v = avg @ Wv^T + bv  and  k = avg @ Wk^T + bk  via V_WMMA_F32_16X16X4_F32
//    grid = HEADS*9 blocks of 32 threads (1 wave).  tile 0..7 -> Wv N-tiles,
//    tile 8 -> Wk (N padded 8->16).  A = avg (8 valid rows, padded to 16).
//    A layout (f32 16x4): lanes 0-15 hold K={0,1}, lanes 16-31 hold K={2,3}.
//    D layout: lanes 0-15, VGPR j -> row M=j (exactly the 8 valid batches).
// ---------------------------------------------------------------------------
__global__ void kv_wmma_kernel(const float* __restrict__ avg,
                               const float* __restrict__ Wv, const float* __restrict__ bv,
                               const float* __restrict__ Wk, const float* __restrict__ bk,
                               float* __restrict__ vout, float* __restrict__ kout) {
    const int h    = blockIdx.x / 9;
    const int tile = blockIdx.x % 9;
    const int lane = threadIdx.x;          // 0..31
    const int m     = lane & 15;           // A row / B-D column within tile
    const int khalf = (lane >> 4) * 2;     // 0 or 2

    v8f acc = {};
    for (int k0 = 0; k0 < HDIM; k0 += 4) {
        v2f a; a.x = 0.f; a.y = 0.f;
        if (m < BATCH) {
            const float* ap = avg + ((size_t)m * HEADS + h) * HDIM + k0 + khalf;
            a.x = ap[0]; a.y = ap[1];
        }
        v2f b; b.x = 0.f; b.y = 0.f;
        if (tile < 8) {
            const int n = tile * 16 + m;   // output feature e
            const float* bp = Wv + ((size_t)h * HDIM + n) * HDIM + k0 + khalf;
            b.x = bp[0]; b.y = bp[1];
        } else if (m < RED) {
            const float* bp = Wk + ((size_t)h * RED + m) * HDIM + k0 + khalf;
            b.x = bp[0]; b.y = bp[1];
        }
        acc = __builtin_amdgcn_wmma_f32_16x16x4_f32(
            /*neg_a=*/false, a, /*neg_b=*/false, b,
            /*c_mod=*/(short)0, acc, /*reuse_a=*/false, /*reuse_b=*/false);
    }

    if (lane < 16) {
        if (tile < 8) {
            const int e = tile * 16 + lane;
            const float bias = bv[h * HDIM + e];
            for (int j = 0; j < BATCH; ++j)      // VGPR j == batch row j
                vout[((size_t)j * HEADS + h) * HDIM + e] = acc[j] + bias;
        } else if (lane < RED) {
            const float bias = bk[h * RED + lane];
            for (int j = 0; j < BATCH; ++j)
                kout[((size_t)j * HEADS + h) * RED + lane] = acc[j] + bias;
        }
    }
}

// ---------------------------------------------------------------------------
// 3) w_eff[b,h,d] = sum_r k[b,h,r]*Wq[h,r,d];  c[b,h] = sum_r k*bq
// ---------------------------------------------------------------------------
__global__ void weff_kernel(const float* __restrict__ kbuf,
                            const float* __restrict__ Wq, const float* __restrict__ bq,
                            float* __restrict__ weff, float* __restrict__ cbuf) {
    const int bh = blockIdx.x;             // b*HEADS + h
    const int h  = bh & (HEADS - 1);
    __shared__ float kv[RED];
    if (threadIdx.x < RED) kv[threadIdx.x] = kbuf[bh * RED + threadIdx.x];
    __syncthreads();
    const int d = threadIdx.x;             // blockDim.x == 128
    float s = 0.f;
    for (int r = 0; r < RED; ++r) s += kv[r] * Wq[((size_t)h * RED + r) * HDIM + d];
    weff[(size_t)bh * HDIM + d] = s;
    if (threadIdx.x == 0) {
        float cc = 0.f;
        for (int r = 0; r < RED; ++r) cc += kv[r] * bq[h * RED + r];
        cbuf[bh] = cc;
    }
}

// ---------------------------------------------------------------------------
// 4) scores[b,h,s] = sum_d x[b, h*128+d, s] * w_eff[d] + c
//    grid = 64*(SPA/512); block owns a 512-wide s-chunk, streams 128 rows.
// ---------------------------------------------------------------------------
__global__ void scores_kernel(const float* __restrict__ x,
                              const float* __restrict__ weff,
                              const float* __restrict__ cbuf,
                              float* __restrict__ scores) {
    const int bh    = blockIdx.x >> 3;     // b*HEADS + h
    const int chunk = blockIdx.x & 7;
    const int b = bh >> 3, h = bh & 7;
    __shared__ float w[HDIM];
    if (threadIdx.x < HDIM) w[threadIdx.x] = weff[(size_t)bh * HDIM + threadIdx.x];
    __syncthreads();
    const float* xb = x + ((size_t)b * CCH + h * HDIM) * SPA + chunk * 512;
    float a0 = 0.f, a1 = 0.f;
    for (int d = 0; d < HDIM; ++d) {
        const float2 xv = ((const float2*)(xb + (size_t)d * SPA))[threadIdx.x];
        a0 += xv.x * w[d];
        a1 += xv.y * w[d];
    }
    const float c = cbuf[bh];
    float2 r; r.x = a0 + c; r.y = a1 + c;
    ((float2*)(scores + (size_t)bh * SPA + chunk * 512))[threadIdx.x] = r;
}

// ---------------------------------------------------------------------------
// 5) per-(b,h): max and 1/sum(exp) over S=4096
// ---------------------------------------------------------------------------
__global__ void stats_kernel(const float* __restrict__ scores, float2* __restrict__ stats) {
    const int bh = blockIdx.x;
    const float4* sp = (const float4*)(scores + (size_t)bh * SPA);
    float4 loc[4];
    float mx = -3.402823466e+38f;
    for (int i = 0; i < 4; ++i) {
        loc[i] = sp[threadIdx.x + i * 256];
        mx = fmaxf(mx, fmaxf(fmaxf(loc[i].x, loc[i].y), fmaxf(loc[i].z, loc[i].w)));
    }
    for (int off = 16; off > 0; off >>= 1) mx = fmaxf(mx, __shfl_xor(mx, off, 32));
    __shared__ float wred[8];
    __shared__ float bmax, bsum;
    const int wid = threadIdx.x >> 5, lane = threadIdx.x & 31;
    if (lane == 0) wred[wid] = mx;
    __syncthreads();
    if (threadIdx.x == 0) {
        float m = wred[0];
        for (int i = 1; i < 8; ++i) m = fmaxf(m, wred[i]);
        bmax = m;
    }
    __syncthreads();
    mx = bmax;
    float s = 0.f;
    for (int i = 0; i < 4; ++i)
        s += __expf(loc[i].x - mx) + __expf(loc[i].y - mx) +
             __expf(loc[i].z - mx) + __expf(loc[i].w - mx);
    for (int off = 16; off > 0; off >>= 1) s += __shfl_xor(s, off, 32);
    __syncthreads();
    if (lane == 0) wred[wid] = s;
    __syncthreads();
    if (threadIdx.x == 0) {
        float t = 0.f;
        for (int i = 0; i < 8; ++i) t += wred[i];
        bsum = t;
        float2 st; st.x = mx; st.y = 1.0f / bsum;
        stats[bh] = st;
    }
}

// ---------------------------------------------------------------------------
// 6) out[b, h*128+e, s] = exp(score-mx)*inv_sum * v[b,h,e]
//    grid = B*CCH; score row stays in L2, reused 128x per (b,h).
// ---------------------------------------------------------------------------
__global__ void out_kernel(const float* __restrict__ scores,
                           const float2* __restrict__ stats,
                           const float* __restrict__ vbuf,
                           float* __restrict__ out) {
    const int row = blockIdx.x;            // b*CCH + h*HDIM + e
    const int b = row >> 10;
    const int he = row & (CCH - 1);
    const int h = he >> 7, e = he & (HDIM - 1);
    const int bh = b * HEADS + h;
    const float2 st = stats[bh];
    const float scale = vbuf[(size_t)bh * HDIM + e] * st.y;
    const float mx = st.x;
    const float4* sp = (const float4*)(scores + (size_t)bh * SPA);
    float4* op = (float4*)(out + (size_t)row * SPA);
    for (int i = threadIdx.x; i < SPA / 4; i += 256) {
        float4 s4 = sp[i];
        float4 o;
        o.x = __expf(s4.x - mx) * scale;
        o.y = __expf(s4.y - mx) * scale;
        o.z = __expf(s4.z - mx) * scale;
        o.w = __expf(s4.w - mx) * scale;
        op[i] = o;
    }
}

// ---------------------------------------------------------------------------
extern "C" void kernel_launch(void* const* d_in, const int* in_sizes, int n_in,
                              void* d_out, int out_size, void* d_ws, size_t ws_size,
                              hipStream_t stream) {
    const float* x  = (const float*)d_in[0];
    const float* Wq = (const float*)d_in[1];
    const float* bq = (const float*)d_in[2];
    const float* Wk = (const float*)d_in[3];
    const float* bk = (const float*)d_in[4];
    const float* Wv = (const float*)d_in[5];
    const float* bv = (const float*)d_in[6];
    float* out = (float*)d_out;

    // workspace layout (16B aligned)
    char* w = (char*)d_ws;
    float*  avg    = (float*)(w);                    // 8*1024           = 32 KB
    float*  kbuf   = (float*)(w + (32 << 10));       // 8*8*8            =  2 KB
    float*  vbuf   = (float*)(w + (36 << 10));       // 8*8*128          = 32 KB
    float*  weff   = (float*)(w + (68 << 10));       // 8*8*128          = 32 KB
    float*  cbuf   = (float*)(w + (100 << 10));      // 64               = 256 B
    float2* stats  = (float2*)(w + (101 << 10));     // 64*8             = 512 B
    float*  scores = (float*)(w + (102 << 10));      // 8*8*4096         =  1 MB

    avg_kernel   <<<BATCH * CCH,        256, 0, stream>>>(x, avg);
    kv_wmma_kernel<<<HEADS * 9,          32, 0, stream>>>(avg, Wv, bv, Wk, bk, vbuf, kbuf);
    weff_kernel  <<<BATCH * HEADS,      128, 0, stream>>>(kbuf, Wq, bq, weff, cbuf);
    scores_kernel<<<BATCH * HEADS * 8,  256, 0, stream>>>(x, weff, cbuf, scores);
    stats_kernel <<<BATCH * HEADS,      256, 0, stream>>>(scores, stats);
    out_kernel   <<<BATCH * CCH,        256, 0, stream>>>(scores, stats, vbuf, out);
}